// ElectricField_76630806495790
// MI455X (gfx1250) — compile-verified
//
#include <hip/hip_runtime.h>
#include <hip/hip_bf16.h>
#include <stdint.h>

// ---------------------------------------------------------------------------
// ElectricField (damped Coulomb field, per-edge scatter-add) — CDNA5 / gfx1250
//
//   uij = d * (P_src*P_dst)^(-1/6)            (BOHR cancels inside uij)
//   contribution on RAW vec: scale = -q[dst] * damp(uij) * BOHR^2 / d^3
//
// Edge stream (153.6 MB) is double-buffered through LDS with
// global_load_async_to_lds_b128 (ASYNCcnt path); node table is an L2-resident
// packed {pol^(-1/6), q} float2; scatter via global_atomic_add_f32.
// ---------------------------------------------------------------------------

typedef float f32x4 __attribute__((ext_vector_type(4)));
typedef float f32x2 __attribute__((ext_vector_type(2)));
typedef int   i32x4 __attribute__((ext_vector_type(4)));

#define DAMP_K (-0.7f * 1.4426950408889634f)   // fold ln->log2 for v_exp_f32
#define BOHR2  ((float)(0.52917721067121 * 0.52917721067121))

#define TILE_EDGES 1024
#define IDX_BYTES  4096                        // 1024 * 4B
#define VEC_BYTES  12288                       // 1024 * 12B
#define BUF_BYTES  24576                       // 3*4096 + 12288
#define TPB        4                           // tiles per block

__global__ void ef_zero_kernel(float* __restrict__ out, int n) {
    int i = blockIdx.x * blockDim.x + threadIdx.x;
    if (i < n) out[i] = 0.0f;
}

__global__ void ef_node_table_kernel(const float* __restrict__ pol,
                                     const float* __restrict__ chg,
                                     f32x2* __restrict__ tbl, int n) {
    int i = blockIdx.x * blockDim.x + threadIdx.x;
    if (i < n) {
        float l2 = __builtin_amdgcn_logf(pol[i]);                   // v_log_f32
        float invp6 = __builtin_amdgcn_exp2f(l2 * (-1.0f / 6.0f));  // v_exp_f32
        f32x2 v;
        v.x = invp6;
        v.y = chg[i];
        tbl[i] = v;
    }
}

__device__ __forceinline__ void ef_edge_body(int si, int di, float dd,
                                             float vx, float vy, float vz,
                                             const f32x2* __restrict__ tbl,
                                             float* __restrict__ out) {
    f32x2 nd = tbl[di];                        // {invp6_dst, q_dst}
    f32x2 ns = tbl[si];                        // {invp6_src, -}
    float u = dd * nd.x * ns.x;
    float arg = DAMP_K * u * __builtin_amdgcn_sqrtf(u);   // -0.7*log2e*u^1.5
    float damp = 1.0f - __builtin_amdgcn_exp2f(arg);
    float inv_d3 = __builtin_amdgcn_rcpf(dd * dd * dd);
    float scale = -nd.y * damp * BOHR2 * inv_d3;
    atomicAdd(&out[3 * si + 0], scale * vx);
    atomicAdd(&out[3 * si + 1], scale * vy);
    atomicAdd(&out[3 * si + 2], scale * vz);
}

// One tile = 1024 edges: issue 6 async b128 copies (each moves 4KB across the
// 256-thread block). offset: is added to BOTH the LDS and global addresses,
// so base pointers for dst/dist/vec are pre-biased by the caller.
__device__ __forceinline__ void ef_issue_tile(unsigned lv, unsigned gv, unsigned gv3,
                                              const void* src_p, const void* dst_m4k,
                                              const void* dist_m8k, const void* vec_m12k) {
    asm volatile("global_load_async_to_lds_b128 %0, %1, %2 th:TH_LOAD_NT"
                 :: "v"(lv), "v"(gv), "s"(src_p) : "memory");
    asm volatile("global_load_async_to_lds_b128 %0, %1, %2 offset:4096 th:TH_LOAD_NT"
                 :: "v"(lv), "v"(gv), "s"(dst_m4k) : "memory");
    asm volatile("global_load_async_to_lds_b128 %0, %1, %2 offset:8192 th:TH_LOAD_NT"
                 :: "v"(lv), "v"(gv), "s"(dist_m8k) : "memory");
    asm volatile("global_load_async_to_lds_b128 %0, %1, %2 offset:12288 th:TH_LOAD_NT"
                 :: "v"(lv), "v"(gv3), "s"(vec_m12k) : "memory");
    asm volatile("global_load_async_to_lds_b128 %0, %1, %2 offset:16384 th:TH_LOAD_NT"
                 :: "v"(lv), "v"(gv3), "s"(vec_m12k) : "memory");
    asm volatile("global_load_async_to_lds_b128 %0, %1, %2 offset:20480 th:TH_LOAD_NT"
                 :: "v"(lv), "v"(gv3), "s"(vec_m12k) : "memory");
}

__global__ void __launch_bounds__(256)
ef_edge_async_kernel(const int* __restrict__ esrc,
                     const void* __restrict__ edst_m4k,
                     const void* __restrict__ dist_m8k,
                     const void* __restrict__ vec_m12k,
                     const int* __restrict__ edst,      // unbiased, for tail
                     const float* __restrict__ dist,
                     const float* __restrict__ vec,
                     const f32x2* __restrict__ tbl,
                     float* __restrict__ out,
                     int n_tiles, int n_edges) {
    __shared__ __align__(16) unsigned char smem[2 * BUF_BYTES];
    const int tid = threadIdx.x;
    const unsigned lane16 = (unsigned)tid * 16u;
    const unsigned lds0 = (unsigned)(uintptr_t)(&smem[0]);

    int t0 = blockIdx.x * TPB;
    int t1 = t0 + TPB;
    if (t1 > n_tiles) t1 = n_tiles;

    if (t0 < t1) {
        // prologue: fill buffer 0 with tile t0
        ef_issue_tile(lds0 + lane16,
                      (unsigned)t0 * (unsigned)IDX_BYTES + lane16,
                      (unsigned)t0 * (unsigned)VEC_BYTES + lane16,
                      esrc, edst_m4k, dist_m8k, vec_m12k);
        int buf = 0;
        for (int t = t0; t < t1; ++t) {
            int tn = t + 1;
            if (tn < t1) {
                unsigned lv = lds0 + (unsigned)((buf ^ 1) * BUF_BYTES) + lane16;
                ef_issue_tile(lv,
                              (unsigned)tn * (unsigned)IDX_BYTES + lane16,
                              (unsigned)tn * (unsigned)VEC_BYTES + lane16,
                              esrc, edst_m4k, dist_m8k, vec_m12k);
                // previous 6 asyncs (current buffer) complete in order
                asm volatile("s_wait_asynccnt 0x6" ::: "memory");
            } else {
                asm volatile("s_wait_asynccnt 0x0" ::: "memory");
            }
            __syncthreads();   // all waves' fills of `buf` visible block-wide

            const unsigned char* b = smem + buf * BUF_BYTES;
            i32x4 s = *(const i32x4*)(b + lane16);
            i32x4 d = *(const i32x4*)(b + IDX_BYTES + lane16);
            f32x4 r = *(const f32x4*)(b + 2 * IDX_BYTES + lane16);
            const f32x4* vp = (const f32x4*)(b + 3 * IDX_BYTES + tid * 48);
            f32x4 v0 = vp[0];
            f32x4 v1 = vp[1];
            f32x4 v2 = vp[2];
            ef_edge_body(s.x, d.x, r.x, v0.x, v0.y, v0.z, tbl, out);
            ef_edge_body(s.y, d.y, r.y, v0.w, v1.x, v1.y, tbl, out);
            ef_edge_body(s.z, d.z, r.z, v1.z, v1.w, v2.x, tbl, out);
            ef_edge_body(s.w, d.w, r.w, v2.y, v2.z, v2.w, tbl, out);

            __syncthreads();   // WAR: done reading `buf` before it is refilled
            buf ^= 1;
        }
    }

    // tail edges (n_tiles*1024 .. n_edges), demand loads, block 0 only
    if (blockIdx.x == 0) {
        for (int e = n_tiles * TILE_EDGES + tid; e < n_edges; e += (int)blockDim.x) {
            ef_edge_body(esrc[e], edst[e], dist[e],
                         vec[3 * (long)e + 0], vec[3 * (long)e + 1],
                         vec[3 * (long)e + 2], tbl, out);
        }
    }
}

extern "C" void kernel_launch(void* const* d_in, const int* in_sizes, int n_in,
                              void* d_out, int out_size, void* d_ws, size_t ws_size,
                              hipStream_t stream) {
    // input order: species, edge_src, edge_dst, distances, vec, charges, polarizability
    const int*   edge_src = (const int*)d_in[1];
    const int*   edge_dst = (const int*)d_in[2];
    const float* dist     = (const float*)d_in[3];
    const float* vec      = (const float*)d_in[4];
    const float* charges  = (const float*)d_in[5];
    const float* pol      = (const float*)d_in[6];
    float*       out      = (float*)d_out;

    const int n_nodes = in_sizes[0];
    const int n_edges = in_sizes[1];

    f32x2* tbl = (f32x2*)d_ws;  // n_nodes * 8B (800 KB for N=100k)

    const int B = 256;  // 8 wave32 waves per block

    ef_zero_kernel<<<(out_size + B - 1) / B, B, 0, stream>>>(out, out_size);
    ef_node_table_kernel<<<(n_nodes + B - 1) / B, B, 0, stream>>>(pol, charges, tbl, n_nodes);

    // pre-biased base pointers: offset: imm is added to BOTH lds & global addr
    const void* edst_m4k  = (const void*)((const char*)edge_dst - IDX_BYTES);
    const void* dist_m8k  = (const void*)((const char*)dist - 2 * IDX_BYTES);
    const void* vec_m12k  = (const void*)((const char*)vec - 3 * IDX_BYTES);

    int n_tiles = n_edges / TILE_EDGES;
    int blocks  = (n_tiles + TPB - 1) / TPB;
    if (blocks < 1) blocks = 1;   // tail-only case still needs one block
    ef_edge_async_kernel<<<blocks, B, 0, stream>>>(
        edge_src, edst_m4k, dist_m8k, vec_m12k,
        edge_dst, dist, vec, tbl, out, n_tiles, n_edges);
}